// ContinualForecaster_12841952215197
// MI455X (gfx1250) — compile-verified
//
#include <hip/hip_runtime.h>
#include <hip/hip_bf16.h>
#include <math.h>

// ---------------- problem constants ----------------
#define B_   8
#define T_   256
#define DI_  64
#define D_   128
#define BT_  (B_*T_)          // 2048
#define PRED_ 96
#define ODIM_ 7
#define OUTD_ (PRED_*ODIM_)   // 672
#define LN_EPS_ 1e-5f

typedef float v2f __attribute__((ext_vector_type(2)));
typedef float v8f __attribute__((ext_vector_type(8)));

// ---------------- math helpers ----------------
__device__ __forceinline__ float gelu_f(float x) {
    // tanh approximation (jax.nn.gelu default approximate=True)
    float x3 = x * x * x;
    float t  = tanhf(0.7978845608028654f * (x + 0.044715f * x3));
    return 0.5f * x * (1.0f + t);
}
__device__ __forceinline__ float sigmoid_f(float x) {
    return 1.0f / (1.0f + expf(-x));
}

// =====================================================================
// Kernel 1: f = gelu(x @ W_b + b_b)    x:(BT,DI) W_b:(DI,D) -> f:(BT,D)
// One wave per 16x16 output tile; fp32 WMMA, K stepped by 4.
// =====================================================================
__global__ __launch_bounds__(128)
void proj_f_kernel(const float* __restrict__ x, const float* __restrict__ Wb,
                   const float* __restrict__ bb, float* __restrict__ f)
{
    const int wave = (blockIdx.x * blockDim.x + threadIdx.x) >> 5;
    const int lane = threadIdx.x & 31;
    const int ntn  = D_ / 16;                 // 8 tiles along N
    const int m0   = (wave / ntn) * 16;
    const int n0   = (wave % ntn) * 16;
    const int rsel = lane & 15;               // row (A) / col (B,C)
    const int kh   = lane >> 4;               // K-half select

    v8f c = {};
#pragma unroll
    for (int k0 = 0; k0 < DI_; k0 += 4) {
        v2f a, b;
        const float* ap = x  + (m0 + rsel) * DI_ + (k0 + 2 * kh);
        a.x = ap[0];                          // K = k0+2*kh
        a.y = ap[1];                          // K = k0+2*kh+1
        const float* bp = Wb + (k0 + 2 * kh) * D_ + (n0 + rsel);
        b.x = bp[0];                          // row K = k0+2*kh
        b.y = bp[D_];                         // row K = k0+2*kh+1
        c = __builtin_amdgcn_wmma_f32_16x16x4_f32(false, a, false, b,
                                                  (short)0, c, false, false);
    }
    const float bias = bb[n0 + rsel];
#pragma unroll
    for (int i = 0; i < 8; ++i) {
        int row = m0 + i + 8 * kh;
        f[row * D_ + n0 + rsel] = gelu_f(c[i] + bias);
    }
}

// =====================================================================
// Kernel 2: k/v/q = f @ {Wk,Wv,Wq}    f:(BT,D)  W:(D,D)
// =====================================================================
__global__ __launch_bounds__(128)
void proj_kvq_kernel(const float* __restrict__ f,
                     const float* __restrict__ Wk, const float* __restrict__ Wv,
                     const float* __restrict__ Wq,
                     float* __restrict__ ko, float* __restrict__ vo,
                     float* __restrict__ qo)
{
    const int wave  = (blockIdx.x * blockDim.x + threadIdx.x) >> 5;
    const int lane  = threadIdx.x & 31;
    const int tiles = (BT_ / 16) * (D_ / 16);     // 1024 tiles per matrix
    const int which = wave / tiles;
    const int tile  = wave % tiles;
    const int ntn   = D_ / 16;
    const int m0    = (tile / ntn) * 16;
    const int n0    = (tile % ntn) * 16;
    const int rsel  = lane & 15;
    const int kh    = lane >> 4;

    const float* W   = (which == 0) ? Wk : (which == 1) ? Wv : Wq;
    float*       out = (which == 0) ? ko : (which == 1) ? vo : qo;

    v8f c = {};
#pragma unroll
    for (int k0 = 0; k0 < D_; k0 += 4) {
        v2f a, b;
        const float* ap = f + (m0 + rsel) * D_ + (k0 + 2 * kh);
        a.x = ap[0];
        a.y = ap[1];
        const float* bp = W + (k0 + 2 * kh) * D_ + (n0 + rsel);
        b.x = bp[0];
        b.y = bp[D_];
        c = __builtin_amdgcn_wmma_f32_16x16x4_f32(false, a, false, b,
                                                  (short)0, c, false, false);
    }
#pragma unroll
    for (int i = 0; i < 8; ++i) {
        int row = m0 + i + 8 * kh;
        out[row * D_ + n0 + rsel] = c[i];
    }
}

// =====================================================================
// Kernel 3: meta = sigmoid(f @ W_m + b_m) -> theta (=0.01*m0), eta, alpha(=0.1*m2)
// =====================================================================
__global__ __launch_bounds__(256)
void meta_kernel(const float* __restrict__ f, const float* __restrict__ Wm,
                 const float* __restrict__ bm,
                 float* __restrict__ theta, float* __restrict__ eta,
                 float* __restrict__ alpha)
{
    int idx = blockIdx.x * blockDim.x + threadIdx.x;
    if (idx >= BT_ * 3) return;
    int row = idx / 3;
    int j   = idx % 3;
    float s = bm[j];
    const float* fr = f + row * D_;
#pragma unroll 8
    for (int i = 0; i < D_; ++i) s += fr[i] * Wm[i * 3 + j];
    float sg = sigmoid_f(s);
    if (j == 0)      theta[row] = 0.01f * sg;
    else if (j == 1) eta[row]   = sg;
    else             alpha[row] = 0.1f * sg;
}

// =====================================================================
// Kernel 4: sequential scan.  One block per batch; M,S (128x128) live in
// registers: thread (r = tid>>2, quarter = tid&3) owns 32 columns of row r.
//  - k_{t+1}/v_{t+1} prefetched into registers while step t computes
//  - row-dot combined with 2x shfl_xor (lanes 4r..4r+3 share a wave32)
//  - per-step scalars are block-uniform -> scalar loads via constant cache
// Only m at t = T-1 is needed downstream -> computed once at the end.
// =====================================================================
__global__ __launch_bounds__(512)
void scan_kernel(const float* __restrict__ k, const float* __restrict__ v,
                 const float* __restrict__ q,
                 const float* __restrict__ theta, const float* __restrict__ eta,
                 const float* __restrict__ alpha,
                 float* __restrict__ m_out)
{
    const int b   = blockIdx.x;
    const int tid = threadIdx.x;
    const int r   = tid >> 2;          // row 0..127
    const int qd  = tid & 3;           // quarter
    const int cb  = qd * 32;           // first owned column

    __shared__ float kbuf[D_];
    __shared__ float vbuf[D_];

    float M[32], S[32];
#pragma unroll
    for (int j = 0; j < 32; ++j) { M[j] = 0.0f; S[j] = 0.0f; }

    const int base = b * T_;

    // prologue: prefetch t = 0
    float kpre = 0.0f, vpre = 0.0f;
    if (tid < D_)           kpre = k[(base + 0) * D_ + tid];
    else if (tid < 2 * D_)  vpre = v[(base + 0) * D_ + (tid - D_)];

    for (int t = 0; t < T_; ++t) {
        // stage current step's k/v from prefetch registers into LDS
        if (tid < D_)           kbuf[tid]      = kpre;
        else if (tid < 2 * D_)  vbuf[tid - D_] = vpre;
        __syncthreads();

        // issue next step's prefetch; completes under the compute below
        if (t + 1 < T_) {
            if (tid < D_)           kpre = k[(base + t + 1) * D_ + tid];
            else if (tid < 2 * D_)  vpre = v[(base + t + 1) * D_ + (tid - D_)];
        }

        // block-uniform per-step scalars (scalar loads)
        const float th  = theta[base + t];
        const float et  = eta[base + t];
        const float oma = 1.0f - alpha[base + t];

        // partial dot over owned 32 columns of row r
        float acc = 0.0f;
#pragma unroll
        for (int j = 0; j < 32; ++j) acc += M[j] * kbuf[cb + j];
        // combine the 4 partials (adjacent lanes of one wave32)
        acc += __shfl_xor(acc, 1, 32);
        acc += __shfl_xor(acc, 2, 32);
        const float e = acc - vbuf[r];         // err[r], in all 4 owner lanes

        // S = eta*S - theta*outer(err,k);  M = (1-alpha)*M + S
#pragma unroll
        for (int j = 0; j < 32; ++j) {
            float g = e * kbuf[cb + j];
            S[j] = et * S[j] - th * g;
            M[j] = oma * M[j] + S[j];
        }
        __syncthreads();   // kbuf/vbuf consumed before next step overwrites
    }

    // m_last = M @ q_{T-1}
    if (tid < D_) kbuf[tid] = q[(base + T_ - 1) * D_ + tid];
    __syncthreads();
    float acc = 0.0f;
#pragma unroll
    for (int j = 0; j < 32; ++j) acc += M[j] * kbuf[cb + j];
    acc += __shfl_xor(acc, 1, 32);
    acc += __shfl_xor(acc, 2, 32);
    if (qd == 0) m_out[b * D_ + r] = acc;
}

// =====================================================================
// Kernel 5: head. One block per batch, 128 threads (thread = feature j).
// gate -> fuse -> W1 -> LayerNorm -> gelu -> W2.
// =====================================================================
__global__ __launch_bounds__(128)
void tail_kernel(const float* __restrict__ f, const float* __restrict__ m_in,
                 const float* __restrict__ Wf, const float* __restrict__ bf,
                 const float* __restrict__ W1, const float* __restrict__ b1,
                 const float* __restrict__ g1, const float* __restrict__ be1,
                 const float* __restrict__ W2, const float* __restrict__ b2,
                 float* __restrict__ out)
{
    const int b = blockIdx.x;
    const int j = threadIdx.x;

    __shared__ float fl[D_];
    __shared__ float ml[D_];
    __shared__ float fu[D_];
    __shared__ float red[D_];

    fl[j] = f[(b * T_ + (T_ - 1)) * D_ + j];
    ml[j] = m_in[b * D_ + j];
    __syncthreads();

    // gate_j = sigmoid( [f,m] @ W_f + b_f )_j
    float a = bf[j];
#pragma unroll 4
    for (int i = 0; i < D_; ++i) a += fl[i] * Wf[i * D_ + j];
#pragma unroll 4
    for (int i = 0; i < D_; ++i) a += ml[i] * Wf[(D_ + i) * D_ + j];
    float g = sigmoid_f(a);
    fu[j] = fl[j] * g + ml[j] * (1.0f - g);
    __syncthreads();

    // h_pre = fused @ W1 + b1
    float hp = b1[j];
#pragma unroll 4
    for (int i = 0; i < D_; ++i) hp += fu[i] * W1[i * D_ + j];

    // LayerNorm over the 128 features
    red[j] = hp;
    __syncthreads();
    for (int s = 64; s > 0; s >>= 1) {
        if (j < s) red[j] += red[j + s];
        __syncthreads();
    }
    float mu = red[0] * (1.0f / D_);
    __syncthreads();
    float d = hp - mu;
    red[j] = d * d;
    __syncthreads();
    for (int s = 64; s > 0; s >>= 1) {
        if (j < s) red[j] += red[j + s];
        __syncthreads();
    }
    float var = red[0] * (1.0f / D_);
    __syncthreads();
    float h = gelu_f(d * rsqrtf(var + LN_EPS_) * g1[j] + be1[j]);

    // reuse fl[] to hold h
    fl[j] = h;
    __syncthreads();

    // out = h @ W2 + b2   (672 outputs; each thread covers j, j+128, ...)
    for (int o = j; o < OUTD_; o += D_) {
        float s = b2[o];
#pragma unroll 4
        for (int i = 0; i < D_; ++i) s += fl[i] * W2[i * OUTD_ + o];
        out[b * OUTD_ + o] = s;
    }
}

// =====================================================================
// launcher
// =====================================================================
extern "C" void kernel_launch(void* const* d_in, const int* in_sizes, int n_in,
                              void* d_out, int out_size, void* d_ws, size_t ws_size,
                              hipStream_t stream)
{
    const float* x   = (const float*)d_in[0];
    const float* Wb  = (const float*)d_in[1];
    const float* bb  = (const float*)d_in[2];
    const float* Wk  = (const float*)d_in[3];
    const float* Wv  = (const float*)d_in[4];
    const float* Wq  = (const float*)d_in[5];
    const float* Wm  = (const float*)d_in[6];
    const float* bm  = (const float*)d_in[7];
    const float* Wf  = (const float*)d_in[8];
    const float* bf  = (const float*)d_in[9];
    const float* W1  = (const float*)d_in[10];
    const float* b1  = (const float*)d_in[11];
    const float* g1  = (const float*)d_in[12];
    const float* be1 = (const float*)d_in[13];
    const float* W2  = (const float*)d_in[14];
    const float* b2  = (const float*)d_in[15];
    float* out = (float*)d_out;

    float* ws    = (float*)d_ws;
    float* f     = ws;                        // BT*D
    float* kbuf  = f     + (size_t)BT_ * D_;  // BT*D
    float* vbuf  = kbuf  + (size_t)BT_ * D_;  // BT*D
    float* qbuf  = vbuf  + (size_t)BT_ * D_;  // BT*D
    float* theta = qbuf  + (size_t)BT_ * D_;  // BT
    float* eta   = theta + BT_;               // BT
    float* alpha = eta   + BT_;               // BT
    float* mlast = alpha + BT_;               // B*D

    // 1) f = gelu(x @ W_b + b_b): 1024 tiles, 4 waves/block -> 256 blocks
    proj_f_kernel<<<256, 128, 0, stream>>>(x, Wb, bb, f);

    // 2) k,v,q: 3*1024 tiles -> 768 blocks
    proj_kvq_kernel<<<768, 128, 0, stream>>>(f, Wk, Wv, Wq, kbuf, vbuf, qbuf);

    // 3) meta: BT*3 = 6144 threads
    meta_kernel<<<(BT_ * 3 + 255) / 256, 256, 0, stream>>>(f, Wm, bm, theta, eta, alpha);

    // 4) sequential scan, one block per batch
    scan_kernel<<<B_, 512, 0, stream>>>(kbuf, vbuf, qbuf, theta, eta, alpha, mlast);

    // 5) head
    tail_kernel<<<B_, 128, 0, stream>>>(f, mlast, Wf, bf, W1, b1, g1, be1, W2, b2, out);
}